// SDFLoss_55035710931613
// MI455X (gfx1250) — compile-verified
//
#include <hip/hip_runtime.h>
#include <math.h>

#define G 32
#define NPTS (G * G * G)
#define FEPS 1e-12f

typedef float v2f __attribute__((ext_vector_type(2)));
typedef float v8f __attribute__((ext_vector_type(8)));

// Per-triangle descriptor: 32 floats = 128 bytes
//  [ 0..3]  colDE0   (e0, -v0.e0)
//  [ 4..7]  colDE1   (e1, -v0.e1)
//  [ 8..11] colG     (-2 v0, |v0|^2)
//  [12..15] colU2    ray-cast u numerator
//  [16..19] colV2    ray-cast v numerator
//  [20..23] a, b, c, z0
//  [24..27] dz1, dz2, inv_det2 (NaN if degenerate), rdet (NaN if det<=EPS)
//  [28..31] ra, rc, ree2, ee2
#define TRI_F 32

// ---------------------------------------------------------------------------
// K0: bounding boxes -> center, scale, overlap flag (single block, 256 thr)
// ---------------------------------------------------------------------------
__global__ void __launch_bounds__(256) k_setup(const float* __restrict__ hv, int nh,
                                               const float* __restrict__ ov, int no,
                                               float* __restrict__ hdr) {
  const float INF = __builtin_inff();
  float hmn[3] = {INF, INF, INF}, hmx[3] = {-INF, -INF, -INF};
  float omn[3] = {INF, INF, INF}, omx[3] = {-INF, -INF, -INF};
  for (int i = threadIdx.x; i < nh; i += blockDim.x) {
#pragma unroll
    for (int k = 0; k < 3; ++k) {
      float v = hv[3 * i + k];
      hmn[k] = fminf(hmn[k], v);
      hmx[k] = fmaxf(hmx[k], v);
    }
  }
  for (int i = threadIdx.x; i < no; i += blockDim.x) {
#pragma unroll
    for (int k = 0; k < 3; ++k) {
      float v = ov[3 * i + k];
      omn[k] = fminf(omn[k], v);
      omx[k] = fmaxf(omx[k], v);
    }
  }
#pragma unroll
  for (int k = 0; k < 3; ++k) {
    for (int m = 1; m < 32; m <<= 1) {
      hmn[k] = fminf(hmn[k], __shfl_xor(hmn[k], m, 32));
      hmx[k] = fmaxf(hmx[k], __shfl_xor(hmx[k], m, 32));
      omn[k] = fminf(omn[k], __shfl_xor(omn[k], m, 32));
      omx[k] = fmaxf(omx[k], __shfl_xor(omx[k], m, 32));
    }
  }
  __shared__ float red[8][12];
  int w = threadIdx.x >> 5;
  if ((threadIdx.x & 31) == 0) {
#pragma unroll
    for (int k = 0; k < 3; ++k) {
      red[w][k] = hmn[k];
      red[w][3 + k] = hmx[k];
      red[w][6 + k] = omn[k];
      red[w][9 + k] = omx[k];
    }
  }
  __syncthreads();
  if (threadIdx.x == 0) {
    float a[12];
#pragma unroll
    for (int k = 0; k < 12; ++k) a[k] = red[0][k];
    for (int i = 1; i < 8; ++i) {
#pragma unroll
      for (int k = 0; k < 3; ++k) {
        a[k] = fminf(a[k], red[i][k]);
        a[3 + k] = fmaxf(a[3 + k], red[i][3 + k]);
        a[6 + k] = fminf(a[6 + k], red[i][6 + k]);
        a[9 + k] = fmaxf(a[9 + k], red[i][9 + k]);
      }
    }
    float ext = fmaxf(fmaxf(a[3] - a[0], a[4] - a[1]), a[5] - a[2]);
    hdr[0] = 0.5f * (a[0] + a[3]);
    hdr[1] = 0.5f * (a[1] + a[4]);
    hdr[2] = 0.5f * (a[2] + a[5]);
    hdr[3] = 1.2f * 0.5f * ext;  // (1 + SCALE_FACTOR) * 0.5 * max extent
    bool olap = (a[0] <= a[9]) && (a[1] <= a[10]) && (a[2] <= a[11]) &&
                (a[6] <= a[3]) && (a[7] <= a[4]) && (a[8] <= a[5]);
    hdr[4] = olap ? 1.0f : 0.0f;
  }
}

// ---------------------------------------------------------------------------
// K1: per-triangle preprocessing into one 128B descriptor (reciprocals here,
// off the hot path; NaN rdet/inv_det2 encode the degenerate-triangle masks)
// ---------------------------------------------------------------------------
__global__ void __launch_bounds__(256) k_tri(const float* __restrict__ hv,
                                             const int* __restrict__ faces,
                                             int F, int Fpad,
                                             const float* __restrict__ hdr,
                                             float* __restrict__ tri) {
  int t = blockIdx.x * blockDim.x + threadIdx.x;
  if (t >= Fpad) return;
  const float qnan = __int_as_float(0x7fc00000);
  float* d = tri + (size_t)t * TRI_F;
  if (t >= F) {
    // padding triangle: huge distance (colG const = 1e30), never ray-hit
    float4 z4 = make_float4(0.f, 0.f, 0.f, 0.f);
    *(float4*)(d + 0) = z4;
    *(float4*)(d + 4) = z4;
    *(float4*)(d + 8) = make_float4(0.f, 0.f, 0.f, 1e30f);
    *(float4*)(d + 12) = z4;
    *(float4*)(d + 16) = z4;
    *(float4*)(d + 20) = z4;                                  // a,b,c,z0
    *(float4*)(d + 24) = make_float4(0.f, 0.f, qnan, qnan);   // dz1,dz2,inv2,rdet
    *(float4*)(d + 28) = make_float4(1.f, 1.f, 1.f, 0.f);    // ra,rc,ree2,ee2
    return;
  }
  float cx = hdr[0], cy = hdr[1], cz = hdr[2];
  float rs = 1.0f / hdr[3];
  int i0 = faces[3 * t], i1 = faces[3 * t + 1], i2 = faces[3 * t + 2];
  float x0 = (hv[3 * i0 + 0] - cx) * rs, y0 = (hv[3 * i0 + 1] - cy) * rs, z0 = (hv[3 * i0 + 2] - cz) * rs;
  float x1 = (hv[3 * i1 + 0] - cx) * rs, y1 = (hv[3 * i1 + 1] - cy) * rs, z1 = (hv[3 * i1 + 2] - cz) * rs;
  float x2 = (hv[3 * i2 + 0] - cx) * rs, y2 = (hv[3 * i2 + 1] - cy) * rs, z2 = (hv[3 * i2 + 2] - cz) * rs;
  float e0x = x1 - x0, e0y = y1 - y0, e0z = z1 - z0;
  float e1x = x2 - x0, e1y = y2 - y0, e1z = z2 - z0;
  float a = e0x * e0x + e0y * e0y + e0z * e0z;
  float b = e0x * e1x + e0y * e1y + e0z * e1z;
  float c = e1x * e1x + e1y * e1y + e1z * e1z;
  // homogeneous linear forms over (px,py,pz,1)
  *(float4*)(d + 0) = make_float4(e0x, e0y, e0z, -(x0 * e0x + y0 * e0y + z0 * e0z));
  *(float4*)(d + 4) = make_float4(e1x, e1y, e1z, -(x0 * e1x + y0 * e1y + z0 * e1z));
  *(float4*)(d + 8) = make_float4(-2.f * x0, -2.f * y0, -2.f * z0,
                                  x0 * x0 + y0 * y0 + z0 * z0);
  float det2 = e0x * e1y - e1x * e0y;
  *(float4*)(d + 12) = make_float4(e1y, -e1x, 0.f, y0 * e1x - x0 * e1y);
  *(float4*)(d + 16) = make_float4(-e0y, e0x, 0.f, x0 * e0y - y0 * e0x);
  float inv2 = (fabsf(det2) > FEPS) ? (1.0f / det2) : qnan;  // NaN => no ray hit
  float det = a * c - b * b;
  float rdet = (det > FEPS) ? (1.0f / det) : qnan;  // NaN => inside_tri false
  float ra = 1.0f / (a > FEPS ? a : 1.0f);
  float rc = 1.0f / (c > FEPS ? c : 1.0f);
  float ee2 = a + c - 2.0f * b;
  float ree2 = 1.0f / (ee2 > FEPS ? ee2 : 1.0f);
  *(float4*)(d + 20) = make_float4(a, b, c, z0);
  *(float4*)(d + 24) = make_float4(z1 - z0, z2 - z0, inv2, rdet);
  *(float4*)(d + 28) = make_float4(ra, rc, ree2, ee2);
}

// ---------------------------------------------------------------------------
// K2: phi over the 32^3 grid. One wave per 16-point tile (2048 waves).
// WMMA f32 16x16x4: A = 16 points x (x,y,z,1), B = 4 x 16 triangle columns.
// D[r] @ lane L  <->  point (r + 8*(L>=16)), triangle (L%16).
// |p|^2 is fed through the C operand of the colG WMMA (layout matches D).
// ---------------------------------------------------------------------------
__global__ void __launch_bounds__(256) k_phi(const float* __restrict__ tri,
                                             int numTiles,
                                             float* __restrict__ phi) {
  const float step = 2.0f / (float)(G - 1);
  int lane = threadIdx.x & 31;
  int ptile = blockIdx.x * (blockDim.x >> 5) + (threadIdx.x >> 5);  // 0..2047
  bool hi = lane >= 16;
  int lr = lane & 15;

  // ---- A operand: 16x4 fp32, lane L holds M=L%16, K=(L<16)?{0,1}:{2,3}
  int pmA = ptile * 16 + lr;
  float ax = -1.0f + step * (float)(pmA & (G - 1));
  float ay = -1.0f + step * (float)((pmA >> 5) & (G - 1));
  float az = -1.0f + step * (float)(pmA >> 10);
  v2f A;
  A.x = hi ? az : ax;   // K=0 or K=2
  A.y = hi ? 1.0f : ay; // K=1 or K=3 (homogeneous 1)

  // ---- per-accumulator-row point constants (uniform within a lane half)
  v8f pp8;
  float pzr[8];
#pragma unroll
  for (int r = 0; r < 8; ++r) {
    int m = ptile * 16 + r + (hi ? 8 : 0);
    float x = -1.0f + step * (float)(m & (G - 1));
    float y = -1.0f + step * (float)((m >> 5) & (G - 1));
    float z = -1.0f + step * (float)(m >> 10);
    pp8[r] = x * x + y * y + z * z;
    pzr[r] = z;
  }

  float dmin[8];
  int hcnt[8];
#pragma unroll
  for (int r = 0; r < 8; ++r) { dmin[r] = 3.0e38f; hcnt[r] = 0; }

  int coff = hi ? 2 : 0;  // B rows K{0,1} in lanes 0-15, K{2,3} in lanes 16-31
  const float* tp = tri + (size_t)lr * TRI_F;  // per-lane descriptor pointer
  for (int tile = 0; tile < numTiles; ++tile, tp += 16 * TRI_F) {
    // one base address; all 8 loads at immediate offsets (clause-friendly)
    v2f bDE0 = *(const v2f*)(tp + 0 + coff);
    v2f bDE1 = *(const v2f*)(tp + 4 + coff);
    v2f bG   = *(const v2f*)(tp + 8 + coff);
    v2f bU2  = *(const v2f*)(tp + 12 + coff);
    v2f bV2  = *(const v2f*)(tp + 16 + coff);
    float4 s0 = *(const float4*)(tp + 20);
    float4 s1 = *(const float4*)(tp + 24);
    float4 s2 = *(const float4*)(tp + 28);

    v8f z8 = {};
    v8f dDE0 = __builtin_amdgcn_wmma_f32_16x16x4_f32(false, A, false, bDE0, (short)0, z8, false, false);
    v8f dDE1 = __builtin_amdgcn_wmma_f32_16x16x4_f32(false, A, false, bDE1, (short)0, z8, false, false);
    v8f dF   = __builtin_amdgcn_wmma_f32_16x16x4_f32(false, A, false, bG,   (short)0, pp8, false, false);
    v8f dU2  = __builtin_amdgcn_wmma_f32_16x16x4_f32(false, A, false, bU2,  (short)0, z8, false, false);
    v8f dV2  = __builtin_amdgcn_wmma_f32_16x16x4_f32(false, A, false, bV2,  (short)0, z8, false, false);

    float a = s0.x, b = s0.y, c = s0.z, z0 = s0.w;
    float dz1 = s1.x, dz2 = s1.y, inv2 = s1.z, rdet = s1.w;
    float ra = s2.x, rc = s2.y, ree2 = s2.z, ee2 = s2.w;
    float bma = b - a;

#pragma unroll
    for (int r = 0; r < 8; ++r) {
      float de0 = dDE0[r], de1 = dDE1[r];
      float f = dF[r];
      // barycentric / interior distance (rdet NaN => inside false)
      float u = (c * de0 - b * de1) * rdet;
      float v = (a * de1 - b * de0) * rdet;
      bool inside = (u >= 0.f) && (v >= 0.f) && (u + v <= 1.f);
      float plane = f - (u * de0 + v * de1);
      // three clamped edge distances
      float t0 = fminf(fmaxf(de0 * ra, 0.f), 1.f);
      float d0 = f + t0 * (t0 * a - 2.0f * de0);
      float t1 = fminf(fmaxf(de1 * rc, 0.f), 1.f);
      float d1 = f + t1 * (t1 * c - 2.0f * de1);
      float dot2 = de1 - de0 - bma;
      float w1sq = f - 2.0f * de0 + a;
      float t2 = fminf(fmaxf(dot2 * ree2, 0.f), 1.f);
      float d2e = w1sq + t2 * (t2 * ee2 - 2.0f * dot2);
      float ed = fminf(fminf(d0, d1), d2e);
      float dd = inside ? plane : ed;
      dd = fmaxf(dd, 0.f);
      dmin[r] = fminf(dmin[r], dd);
      // z-ray parity test (NaN inv2 => no hit)
      float u2 = dU2[r] * inv2;
      float v2b = dV2[r] * inv2;
      bool in2 = (u2 >= 0.f) && (v2b >= 0.f) && ((u2 + v2b) <= 1.f);
      float zint = z0 + u2 * dz1 + v2b * dz2;
      hcnt[r] += (in2 && (zint > pzr[r])) ? 1 : 0;
    }
  }

  // min/sum over the 16 triangle lanes of each half (butterfly all-reduce)
#pragma unroll
  for (int r = 0; r < 8; ++r) {
    float v = dmin[r];
    int h = hcnt[r];
    for (int m = 1; m < 16; m <<= 1) {
      v = fminf(v, __shfl_xor(v, m, 32));
      h += __shfl_xor(h, m, 32);
    }
    if (lr == r) {  // lanes r and 16+r write points r and r+8 of the tile
      int pt = ptile * 16 + r + (hi ? 8 : 0);
      phi[pt] = sqrtf(v) * (float)(h & 1);
    }
  }
}

// ---------------------------------------------------------------------------
// K3: trilinear zero-padded sampling (align_corners=True) + squared-sum loss
// ---------------------------------------------------------------------------
__global__ void __launch_bounds__(256) k_loss(const float* __restrict__ ov, int no,
                                              const float* __restrict__ hdr,
                                              const float* __restrict__ phi,
                                              float* __restrict__ out) {
  float cx = hdr[0], cy = hdr[1], cz = hdr[2];
  float rs = 1.0f / hdr[3];
  float sum = 0.0f;
  for (int i = threadIdx.x; i < no; i += blockDim.x) {
    float qx = (ov[3 * i + 0] - cx) * rs;
    float qy = (ov[3 * i + 1] - cy) * rs;
    float qz = (ov[3 * i + 2] - cz) * rs;
    float fx = (qx + 1.0f) * 0.5f * (float)(G - 1);
    float fy = (qy + 1.0f) * 0.5f * (float)(G - 1);
    float fz = (qz + 1.0f) * 0.5f * (float)(G - 1);
    float flx = floorf(fx), fly = floorf(fy), flz = floorf(fz);
    int ix0 = (int)flx, iy0 = (int)fly, iz0 = (int)flz;
    float wx1 = fx - flx, wy1 = fy - fly, wz1 = fz - flz;
    float wx0 = 1.0f - wx1, wy0 = 1.0f - wy1, wz0 = 1.0f - wz1;
#pragma unroll
    for (int dz = 0; dz < 2; ++dz)
#pragma unroll
      for (int dy = 0; dy < 2; ++dy)
#pragma unroll
        for (int dx = 0; dx < 2; ++dx) {
          int ix = ix0 + dx, iy = iy0 + dy, iz = iz0 + dz;
          if ((unsigned)ix < G && (unsigned)iy < G && (unsigned)iz < G) {
            float val = phi[(iz * G + iy) * G + ix];
            float w = (dx ? wx1 : wx0) * (dy ? wy1 : wy0) * (dz ? wz1 : wz0);
            sum += val * w;
          }
        }
  }
  for (int m = 1; m < 32; m <<= 1) sum += __shfl_xor(sum, m, 32);
  __shared__ float red[8];
  if ((threadIdx.x & 31) == 0) red[threadIdx.x >> 5] = sum;
  __syncthreads();
  if (threadIdx.x == 0) {
    float s = 0.0f;
    for (int i = 0; i < 8; ++i) s += red[i];
    out[0] = (hdr[4] != 0.0f) ? s * s : 0.0f;
  }
}

// ---------------------------------------------------------------------------
extern "C" void kernel_launch(void* const* d_in, const int* in_sizes, int n_in,
                              void* d_out, int out_size, void* d_ws, size_t ws_size,
                              hipStream_t stream) {
  const float* hv = (const float*)d_in[0];
  const float* ov = (const float*)d_in[1];
  const int* faces = (const int*)d_in[2];
  int nh = in_sizes[0] / 3;   // 778
  int no = in_sizes[1] / 3;   // 4096
  int F = in_sizes[2] / 3;    // 1538
  int numTiles = (F + 15) / 16;
  int Fpad = numTiles * 16;

  char* ws = (char*)d_ws;
  float* hdr = (float*)ws;                       // 16 floats
  float* tri = (float*)(ws + 64);                // Fpad * 32 floats (128B each)
  float* phi = tri + (size_t)Fpad * TRI_F;       // NPTS floats

  k_setup<<<1, 256, 0, stream>>>(hv, nh, ov, no, hdr);
  k_tri<<<(Fpad + 255) / 256, 256, 0, stream>>>(hv, faces, F, Fpad, hdr, tri);
  // 2048 point tiles -> 2048 waves -> 256 blocks of 8 waves
  k_phi<<<NPTS / 16 / 8, 256, 0, stream>>>(tri, numTiles, phi);
  k_loss<<<1, 256, 0, stream>>>(ov, no, hdr, phi, (float*)d_out);
}